// WindowTransformerBlock_53360673685511
// MI455X (gfx1250) — compile-verified
//
#include <hip/hip_runtime.h>

namespace {

constexpr int WSZ   = 7;
constexpr int HW    = 200;
constexpr int NWIN  = 29;        // ceil(200/7) -> 29 windows per axis (pad 3)
constexpr int BATCH = 8;
constexpr int CH    = 256;
constexpr int NHEAD = 8;
constexpr int DH    = 32;
constexpr int DFF   = 1024;
constexpr int NTOK  = 49;
constexpr int MPAD  = 64;        // token rows padded to 64 for 16x16 WMMA tiles
constexpr int SST   = 257;       // LDS stride for 64x256 buffers (odd -> conflict free)
constexpr int KST   = 33;        // LDS stride for 64x32 q/k/v
constexpr int PST   = 65;        // LDS stride for 64x64 scores
constexpr int TST   = 129;       // LDS stride for 64x128 FF1 chunk
constexpr int FCH   = 128;       // DFF chunk width

typedef __attribute__((ext_vector_type(2))) float v2f;
typedef __attribute__((ext_vector_type(8))) float v8f;

__device__ __forceinline__ v8f wmma4(v2f a, v2f b, v8f c) {
  // D = A(16x4 f32) * B(4x16 f32) + C(16x16 f32)
  return __builtin_amdgcn_wmma_f32_16x16x4_f32(false, a, false, b, (short)0, c,
                                               false, false);
}

__device__ __forceinline__ float wave_sum(float v) {
#pragma unroll
  for (int o = 16; o > 0; o >>= 1) v += __shfl_xor(v, o, 32);
  return v;
}
__device__ __forceinline__ float wave_max(float v) {
#pragma unroll
  for (int o = 16; o > 0; o >>= 1) v = fmaxf(v, __shfl_xor(v, o, 32));
  return v;
}

// ---- WMMA fragment helpers (f32, wave32 layouts per CDNA5 ISA 7.12.2) ----
// A fragment / transposed-B: element [row0 + (lane&15)][k], k pair selected by lane>=16
__device__ __forceinline__ v2f frag_rowk(const float* buf, int stride, int row0,
                                         int k0, int lane) {
  int r = row0 + (lane & 15);
  int k = k0 + ((lane >> 4) << 1);
  const float* p = buf + r * stride + k;
  v2f f; f.x = p[0]; f.y = p[1];
  return f;
}
// A fragment with LayerNorm applied on the fly (buf has stride SST)
__device__ __forceinline__ v2f frag_ln(const float* buf, const float* lw,
                                       const float* lb, const float* meanS,
                                       const float* rstdS, int row0, int k0,
                                       int lane) {
  int r = row0 + (lane & 15);
  int k = k0 + ((lane >> 4) << 1);
  float m = meanS[r], rs = rstdS[r];
  const float* p = buf + r * SST + k;
  v2f f;
  f.x = (p[0] - m) * rs * lw[k] + lb[k];
  f.y = (p[1] - m) * rs * lw[k + 1] + lb[k + 1];
  return f;
}
// B fragment from a row-major (out x in) global weight: B[k][n] = W[ncol+n][k]
__device__ __forceinline__ v2f frag_wg(const float* __restrict__ W, int ldw,
                                       int ncol, int k0, int lane) {
  int n = ncol + (lane & 15);
  int k = k0 + ((lane >> 4) << 1);
  const float* p = W + (size_t)n * ldw + k;      // k0 even -> 8B aligned
  v2f f; f.x = p[0]; f.y = p[1];
  return f;
}
// B fragment from LDS stored K-major: B[k][n] = buf[k*stride + ncol + n]
__device__ __forceinline__ v2f frag_kn(const float* buf, int stride, int ncol,
                                       int k0, int lane) {
  int n = ncol + (lane & 15);
  int k = k0 + ((lane >> 4) << 1);
  v2f f;
  f.x = buf[k * stride + n];
  f.y = buf[(k + 1) * stride + n];
  return f;
}
// Store D fragment into LDS
__device__ __forceinline__ void frag_store(float* buf, int stride, int row0,
                                           int ncol, v8f c, int lane) {
  int r = row0 + ((lane >> 4) << 3);
  int n = ncol + (lane & 15);
#pragma unroll
  for (int j = 0; j < 8; ++j) buf[(r + j) * stride + n] = c[j];
}

__device__ __forceinline__ void row_stats(const float* sS, float* meanS,
                                          float* rstdS, int wave, int lane) {
  for (int r = wave; r < MPAD; r += 8) {
    float s = 0.f, s2 = 0.f;
    for (int k = lane; k < CH; k += 32) {
      float v = sS[r * SST + k];
      s += v; s2 += v * v;
    }
    s = wave_sum(s); s2 = wave_sum(s2);
    if (lane == 0) {
      float m = s * (1.f / CH);
      float var = fmaxf(s2 * (1.f / CH) - m * m, 0.f);
      meanS[r] = m;
      rstdS[r] = rsqrtf(var + 1e-5f);
    }
  }
}

__global__ __launch_bounds__(256, 1) void win_block_kernel(
    const float* __restrict__ x, const float* __restrict__ pos,
    const float* __restrict__ wqkv, const float* __restrict__ bqkv,
    const float* __restrict__ wo, const float* __restrict__ bo,
    const float* __restrict__ ln1w, const float* __restrict__ ln1b,
    const float* __restrict__ ln2w, const float* __restrict__ ln2b,
    const float* __restrict__ w1, const float* __restrict__ b1,
    const float* __restrict__ w2, const float* __restrict__ b2,
    float* __restrict__ out) {
  __shared__ float sS[MPAD * SST];                       // residual stream s
  __shared__ float oS[MPAD * SST];                       // attn-out, then FF accum
  __shared__ float uS[3 * MPAD * KST + MPAD * PST];      // q,k,v,scores | FF1 chunk
  __shared__ float meanS[MPAD], rstdS[MPAD];
  __shared__ float l1wS[CH], l1bS[CH], l2wS[CH], l2bS[CH];
  __shared__ float bqkvS[3 * CH], b1S[DFF];

  float* qS = uS;
  float* kS = qS + MPAD * KST;
  float* vS = kS + MPAD * KST;
  float* pS = vS + MPAD * KST;
  float* tS = uS;                                        // overlay (attn bufs dead)

  const int tid  = threadIdx.x;
  const int wave = tid >> 5;
  const int lane = tid & 31;

  const int win = blockIdx.x;
  const int b   = win / (NWIN * NWIN);
  const int wr  = (win / NWIN) % NWIN;
  const int wc  = win % NWIN;

  // ---- stage small params in LDS ----
  l1wS[tid] = ln1w[tid]; l1bS[tid] = ln1b[tid];
  l2wS[tid] = ln2w[tid]; l2bS[tid] = ln2b[tid];
  for (int i = tid; i < 3 * CH; i += 256) bqkvS[i] = bqkv[i];
  for (int i = tid; i < DFF; i += 256) b1S[i] = b1[i];

  // ---- window gather + pos embed, zero-pad rows 49..63 ----
  {
    const int c = tid;
    const float* xb = x + ((size_t)b * CH + c) * (HW * HW);
    for (int t = 0; t < NTOK; ++t) {
      int i = t / WSZ, j = t % WSZ;
      int hh = wr * WSZ + i, ww = wc * WSZ + j;
      float v = 0.f;
      if (hh < HW && ww < HW) v = xb[hh * HW + ww];
      sS[t * SST + c] = v + pos[t * CH + c];
    }
    for (int t = NTOK; t < MPAD; ++t) sS[t * SST + c] = 0.f;
  }
  __syncthreads();

  // ---- LN1 row statistics ----
  row_stats(sS, meanS, rstdS, wave, lane);
  __syncthreads();

  const float zero8[8] = {};
  (void)zero8;

  // ================= attention, head by head =================
  for (int h = 0; h < NHEAD; ++h) {
    // q/k/v for this head: (64x256 LN1(s)) @ Wqkv_head^T -> 64x96
    for (int t = wave; t < 24; t += 8) {            // 4 row tiles x 6 col tiles
      int mt = t & 3, cg = t >> 2;
      int part = cg >> 1;                           // 0=q 1=k 2=v
      int wrow = part * CH + h * DH + (cg & 1) * 16;
      v8f acc = {0.f, 0.f, 0.f, 0.f, 0.f, 0.f, 0.f, 0.f};
      for (int k0 = 0; k0 < CH; k0 += 4) {
        v2f a  = frag_ln(sS, l1wS, l1bS, meanS, rstdS, mt * 16, k0, lane);
        v2f bf = frag_wg(wqkv, CH, wrow, k0, lane);
        acc = wmma4(a, bf, acc);
      }
      float bias = bqkvS[wrow + (lane & 15)];
#pragma unroll
      for (int j = 0; j < 8; ++j) acc[j] += bias;
      float* dst = (part == 0) ? qS : (part == 1) ? kS : vS;
      frag_store(dst, KST, mt * 16, (cg & 1) * 16, acc, lane);
    }
    __syncthreads();

    // scores = q @ k^T / sqrt(dh)  (64x64)
    for (int t = wave; t < 16; t += 8) {
      int mt = t & 3, nt = t >> 2;
      v8f acc = {0.f, 0.f, 0.f, 0.f, 0.f, 0.f, 0.f, 0.f};
      for (int k0 = 0; k0 < DH; k0 += 4) {
        v2f a  = frag_rowk(qS, KST, mt * 16, k0, lane);
        v2f bf = frag_rowk(kS, KST, nt * 16, k0, lane);  // B[k][n]=k[n][k]
        acc = wmma4(a, bf, acc);
      }
      const float scale = 0.17677669529663687f;          // 1/sqrt(32)
#pragma unroll
      for (int j = 0; j < 8; ++j) acc[j] *= scale;
      frag_store(pS, PST, mt * 16, nt * 16, acc, lane);
    }
    __syncthreads();

    // softmax over 49 valid keys; zero the padding
    for (int r = wave; r < MPAD; r += 8) {
      if (r < NTOK) {
        float v0 = pS[r * PST + lane];
        float v1 = (lane + 32 < NTOK) ? pS[r * PST + lane + 32] : -3.0e38f;
        float mx = wave_max(fmaxf(v0, v1));
        float e0 = expf(v0 - mx);
        float e1 = (lane + 32 < NTOK) ? expf(v1 - mx) : 0.f;
        float inv = 1.f / wave_sum(e0 + e1);
        pS[r * PST + lane]      = e0 * inv;
        pS[r * PST + lane + 32] = e1 * inv;   // zeros for cols 49..63
      } else {
        pS[r * PST + lane] = 0.f;
        pS[r * PST + lane + 32] = 0.f;
      }
    }
    __syncthreads();

    // o_head = P @ v  -> oS[:, h*32 .. h*32+31]
    {
      int mt = wave & 3, nt = wave >> 2;               // exactly 8 tiles
      v8f acc = {0.f, 0.f, 0.f, 0.f, 0.f, 0.f, 0.f, 0.f};
      for (int k0 = 0; k0 < MPAD; k0 += 4) {
        v2f a  = frag_rowk(pS, PST, mt * 16, k0, lane);
        v2f bf = frag_kn(vS, KST, nt * 16, k0, lane);
        acc = wmma4(a, bf, acc);
      }
      frag_store(oS, SST, mt * 16, h * DH + nt * 16, acc, lane);
    }
    __syncthreads();
  }

  // ---- out projection + residual: s += o @ Wo^T + bo ----
  for (int t = wave; t < 64; t += 8) {
    int mt = t & 3, nt = t >> 2;
    v8f acc = {0.f, 0.f, 0.f, 0.f, 0.f, 0.f, 0.f, 0.f};
    for (int k0 = 0; k0 < CH; k0 += 4) {
      v2f a  = frag_rowk(oS, SST, mt * 16, k0, lane);
      v2f bf = frag_wg(wo, CH, nt * 16, k0, lane);
      acc = wmma4(a, bf, acc);
    }
    float bias = bo[nt * 16 + (lane & 15)];
    int r = mt * 16 + ((lane >> 4) << 3);
    int n = nt * 16 + (lane & 15);
#pragma unroll
    for (int j = 0; j < 8; ++j) sS[(r + j) * SST + n] += acc[j] + bias;
  }
  __syncthreads();

  // ---- LN2 stats ----
  row_stats(sS, meanS, rstdS, wave, lane);
  __syncthreads();

  // ================= FFN, chunked over DFF =================
  for (int chunk = 0; chunk < DFF / FCH; ++chunk) {
    // FF1: t = gelu(LN2(s) @ W1_chunk^T + b1)  (64x128)
    const float* w1c = w1 + (size_t)chunk * FCH * CH;
    for (int t = wave; t < 32; t += 8) {               // 4 x 8 tiles
      int mt = t & 3, nt = t >> 2;
      v8f acc = {0.f, 0.f, 0.f, 0.f, 0.f, 0.f, 0.f, 0.f};
      for (int k0 = 0; k0 < CH; k0 += 4) {
        v2f a  = frag_ln(sS, l2wS, l2bS, meanS, rstdS, mt * 16, k0, lane);
        v2f bf = frag_wg(w1c, CH, nt * 16, k0, lane);
        acc = wmma4(a, bf, acc);
      }
      float bias = b1S[chunk * FCH + nt * 16 + (lane & 15)];
#pragma unroll
      for (int j = 0; j < 8; ++j) {
        float hh = acc[j] + bias;
        acc[j] = 0.5f * hh * (1.f + erff(hh * 0.70710678118654752f));  // exact gelu
      }
      frag_store(tS, TST, mt * 16, nt * 16, acc, lane);
    }
    __syncthreads();

    // FF2 partial: oS (+)= t @ W2[:, chunk]^T   (accumulate via C operand)
    for (int t = wave; t < 64; t += 8) {
      int mt = t & 3, nt = t >> 2;
      v8f acc;
      if (chunk == 0) {
        acc = v8f{0.f, 0.f, 0.f, 0.f, 0.f, 0.f, 0.f, 0.f};
      } else {
        int r = mt * 16 + ((lane >> 4) << 3);
        int n = nt * 16 + (lane & 15);
#pragma unroll
        for (int j = 0; j < 8; ++j) acc[j] = oS[(r + j) * SST + n];
      }
      for (int k0 = 0; k0 < FCH; k0 += 4) {
        v2f a  = frag_rowk(tS, TST, mt * 16, k0, lane);
        v2f bf = frag_wg(w2 + chunk * FCH, DFF, nt * 16, k0, lane);
        acc = wmma4(a, bf, acc);
      }
      frag_store(oS, SST, mt * 16, nt * 16, acc, lane);
    }
    __syncthreads();
  }

  // ---- window reverse + crop: out = s + ff_out + b2 ----
  {
    const int c = tid;
    const float bias2 = b2[c];
    float* ob = out + ((size_t)b * CH + c) * (HW * HW);
    for (int t = 0; t < NTOK; ++t) {
      int i = t / WSZ, j = t % WSZ;
      int hh = wr * WSZ + i, ww = wc * WSZ + j;
      if (hh < HW && ww < HW)
        ob[hh * HW + ww] = sS[t * SST + c] + oS[t * SST + c] + bias2;
    }
  }
}

}  // namespace

extern "C" void kernel_launch(void* const* d_in, const int* in_sizes, int n_in,
                              void* d_out, int out_size, void* d_ws,
                              size_t ws_size, hipStream_t stream) {
  (void)in_sizes; (void)n_in; (void)out_size; (void)d_ws; (void)ws_size;
  const float* x    = (const float*)d_in[0];
  const float* pos  = (const float*)d_in[1];
  const float* wqkv = (const float*)d_in[2];
  const float* bqkv = (const float*)d_in[3];
  const float* wo   = (const float*)d_in[4];
  const float* bo   = (const float*)d_in[5];
  const float* l1w  = (const float*)d_in[6];
  const float* l1b  = (const float*)d_in[7];
  const float* l2w  = (const float*)d_in[8];
  const float* l2b  = (const float*)d_in[9];
  const float* w1   = (const float*)d_in[10];
  const float* b1   = (const float*)d_in[11];
  const float* w2   = (const float*)d_in[12];
  const float* b2   = (const float*)d_in[13];
  float* out = (float*)d_out;

  dim3 grid(BATCH * NWIN * NWIN);   // 6728 windows
  dim3 block(256);                  // 8 wave32
  hipLaunchKernelGGL(win_block_kernel, grid, block, 0, stream,
                     x, pos, wqkv, bqkv, wo, bo, l1w, l1b, l2w, l2b,
                     w1, b1, w2, b2, out);
}